// GroupedBCEWithLogitsLoss_42210938585615
// MI455X (gfx1250) — compile-verified
//
#include <hip/hip_runtime.h>
#include <math.h>

// Problem constants from the reference
#define G_GROUPS 32768      // number of segments (power of two)
#define BLOCK1   1024       // 32 wave32 waves per workgroup (fills a WGP)
#define CBLOCK   256
#define POS_W    1.0f

// Staging for async global->LDS pipeline: 2 buffers x BLOCK1 lanes x 8B x 2 arrays
#define STAGE_BYTES (2 * BLOCK1 * 8 * 2)                 // 32 KB
#define SH_SUMCOUNT (G_GROUPS * 8 + STAGE_BYTES)         // 256 KB + 32 KB = 288 KB
#define SH_MAX      (G_GROUPS * 4 + STAGE_BYTES)         // 128 KB + 32 KB = 160 KB

// ---------------------------------------------------------------------------
// gfx1250 async global<->LDS ops (ASYNCcnt-tracked). The clang builtins take
// pointers to int vectors (vector_size 8/16) in AS(1)/AS(3), per the compiler
// diagnostics. Guarded so the file compiles on any toolchain; host pass sees
// HAVE_ASYNC_LDS==0, which is fine because the dynamic-LDS sizes are fixed.
// ---------------------------------------------------------------------------
#if defined(__AMDGCN__) && \
    __has_builtin(__builtin_amdgcn_global_load_async_to_lds_b64) && \
    __has_builtin(__builtin_amdgcn_global_store_async_from_lds_b128)
#define HAVE_ASYNC_LDS 1
#else
#define HAVE_ASYNC_LDS 0
#endif

#if HAVE_ASYNC_LDS
typedef int v2i_t __attribute__((vector_size(8)));    // 2 x i32
typedef int v4i_t __attribute__((vector_size(16)));   // 4 x i32
typedef __attribute__((address_space(1))) v2i_t* gp2_t;   // global
typedef __attribute__((address_space(3))) v2i_t* lp2_t;   // LDS
typedef __attribute__((address_space(1))) v4i_t* gp4_t;
typedef __attribute__((address_space(3))) v4i_t* lp4_t;

__device__ __forceinline__ void async_ld_b64(const void* g, void* l) {
    // global_load_async_to_lds_b64: per-lane 8B, no VGPR dest, ASYNCcnt-tracked
    __builtin_amdgcn_global_load_async_to_lds_b64((gp2_t)g, (lp2_t)l, 0, 0);
}
__device__ __forceinline__ void async_st_b128(void* g, const void* l) {
    // global_store_async_from_lds_b128: LDS -> HBM without a VGPR round trip
    __builtin_amdgcn_global_store_async_from_lds_b128((gp4_t)g, (lp4_t)l, 0, 0);
}
template <int N> __device__ __forceinline__ void wait_async() {
#if __has_builtin(__builtin_amdgcn_s_wait_asynccnt)
    __builtin_amdgcn_s_wait_asynccnt(N);
#else
    asm volatile("s_wait_asynccnt %0" :: "n"(N) : "memory");
#endif
    asm volatile("" ::: "memory");   // compiler ordering: keep ds_loads below the wait
}
#endif

// Numerically stable log(sigmoid(x)) = min(x,0) - log1p(exp(-|x|))
__device__ __forceinline__ float log_sigmoid_f(float x) {
    return fminf(x, 0.0f) - log1pf(expf(-fabsf(x)));
}

// ---------------------------------------------------------------------------
// Kernel 0: zero the scalar accumulator (d_ws is poisoned, never re-zeroed).
// ---------------------------------------------------------------------------
__global__ void k_init(float* __restrict__ accum) {
    if (threadIdx.x == 0) accum[0] = 0.0f;
}

// ---------------------------------------------------------------------------
// Kernel 1: per-chunk privatized segment sum + count.
// Full G-sized accumulator in LDS (256 KB of the 320 KB CDNA5 WGP LDS).
// Streaming reads arrive via double-buffered async-to-LDS copies; the
// histogram flush leaves via async-from-LDS b128 stores.
// nvec2 = number of float2/int2 vectors in this chunk.
// ---------------------------------------------------------------------------
__global__ void __launch_bounds__(BLOCK1)
k_sum_count(const float* __restrict__ inp, const int* __restrict__ gid,
            float* __restrict__ psum, unsigned* __restrict__ pcnt,
            int nvec2) {
    extern __shared__ char smem[];
    float*    ls = reinterpret_cast<float*>(smem);                      // [G]
    unsigned* lc = reinterpret_cast<unsigned*>(smem + G_GROUPS * 4);    // [G]
    const int tid = threadIdx.x;

    for (int i = tid; i < G_GROUPS; i += BLOCK1) { ls[i] = 0.0f; lc[i] = 0u; }
    __syncthreads();

    const size_t vbase = (size_t)blockIdx.x * (size_t)nvec2;
    const float2* in2 = reinterpret_cast<const float2*>(inp) + vbase;
    const int2*   gd2 = reinterpret_cast<const int2*>(gid) + vbase;

#if HAVE_ASYNC_LDS
    float2* st_in = reinterpret_cast<float2*>(smem + G_GROUPS * 8);            // [2*BLOCK1]
    int2*   st_gd = reinterpret_cast<int2*>(smem + G_GROUPS * 8 + 2 * BLOCK1 * 8);

    const int tiles = nvec2 / BLOCK1;
    if (tiles > 0) {
        async_ld_b64(in2 + tid, st_in + tid);
        async_ld_b64(gd2 + tid, st_gd + tid);
    }
    for (int t = 0; t < tiles; ++t) {
        const int buf = (t & 1) * BLOCK1;
        if (t + 1 < tiles) {
            const int nbuf = ((t + 1) & 1) * BLOCK1;
            async_ld_b64(in2 + (size_t)(t + 1) * BLOCK1 + tid, st_in + nbuf + tid);
            async_ld_b64(gd2 + (size_t)(t + 1) * BLOCK1 + tid, st_gd + nbuf + tid);
            wait_async<2>();   // loads complete in order: tile t's 2 ops are done
        } else {
            wait_async<0>();
        }
        float2 x = st_in[buf + tid];   // ds_load_b64 of own staged data: no barrier needed
        int2   g = st_gd[buf + tid];
        atomicAdd(&ls[g.x & (G_GROUPS - 1)], x.x); atomicAdd(&lc[g.x & (G_GROUPS - 1)], 1u);
        atomicAdd(&ls[g.y & (G_GROUPS - 1)], x.y); atomicAdd(&lc[g.y & (G_GROUPS - 1)], 1u);
    }
    for (int v = tiles * BLOCK1 + tid; v < nvec2; v += BLOCK1) {   // remainder
        float2 x = in2[v]; int2 g = gd2[v];
        atomicAdd(&ls[g.x & (G_GROUPS - 1)], x.x); atomicAdd(&lc[g.x & (G_GROUPS - 1)], 1u);
        atomicAdd(&ls[g.y & (G_GROUPS - 1)], x.y); atomicAdd(&lc[g.y & (G_GROUPS - 1)], 1u);
    }
#else
    for (int v = tid; v < nvec2; v += BLOCK1) {
        __builtin_prefetch(in2 + v + BLOCK1, 0, 0);
        __builtin_prefetch(gd2 + v + BLOCK1, 0, 0);
        float2 x = in2[v]; int2 g = gd2[v];
        atomicAdd(&ls[g.x & (G_GROUPS - 1)], x.x); atomicAdd(&lc[g.x & (G_GROUPS - 1)], 1u);
        atomicAdd(&ls[g.y & (G_GROUPS - 1)], x.y); atomicAdd(&lc[g.y & (G_GROUPS - 1)], 1u);
    }
#endif
    __syncthreads();

    // Flush partials: coalesced, atomic-free.
    float*    os = psum + (size_t)blockIdx.x * G_GROUPS;
    unsigned* oc = pcnt + (size_t)blockIdx.x * G_GROUPS;
#if HAVE_ASYNC_LDS
    float4* os4 = reinterpret_cast<float4*>(os);
    uint4*  oc4 = reinterpret_cast<uint4*>(oc);
    float4* ls4 = reinterpret_cast<float4*>(ls);
    uint4*  lc4 = reinterpret_cast<uint4*>(lc);
    for (int i = tid; i < G_GROUPS / 4; i += BLOCK1) {
        async_st_b128(os4 + i, ls4 + i);
        async_st_b128(oc4 + i, lc4 + i);
    }
    wait_async<0>();
#else
    for (int i = tid; i < G_GROUPS; i += BLOCK1) { os[i] = ls[i]; oc[i] = lc[i]; }
#endif
}

// ---------------------------------------------------------------------------
// Kernel 2: per-chunk privatized segment max of target.
// Targets are uniform[0,1) (non-negative) so IEEE order == int order on bits:
// ds_max_i32 LDS atomics, slots initialized to -inf bits.
// ---------------------------------------------------------------------------
__global__ void __launch_bounds__(BLOCK1)
k_max(const float* __restrict__ tgt, const int* __restrict__ gid,
      float* __restrict__ pmax, int nvec2) {
    extern __shared__ char smem2[];
    int* lm = reinterpret_cast<int*>(smem2);                            // [G]
    const int tid = threadIdx.x;
    const int NEG_INF_BITS = (int)0xFF800000;
    for (int i = tid; i < G_GROUPS; i += BLOCK1) lm[i] = NEG_INF_BITS;
    __syncthreads();

    const size_t vbase = (size_t)blockIdx.x * (size_t)nvec2;
    const float2* tg2 = reinterpret_cast<const float2*>(tgt) + vbase;
    const int2*   gd2 = reinterpret_cast<const int2*>(gid) + vbase;

#if HAVE_ASYNC_LDS
    float2* st_tg = reinterpret_cast<float2*>(smem2 + G_GROUPS * 4);
    int2*   st_gd = reinterpret_cast<int2*>(smem2 + G_GROUPS * 4 + 2 * BLOCK1 * 8);

    const int tiles = nvec2 / BLOCK1;
    if (tiles > 0) {
        async_ld_b64(tg2 + tid, st_tg + tid);
        async_ld_b64(gd2 + tid, st_gd + tid);
    }
    for (int t = 0; t < tiles; ++t) {
        const int buf = (t & 1) * BLOCK1;
        if (t + 1 < tiles) {
            const int nbuf = ((t + 1) & 1) * BLOCK1;
            async_ld_b64(tg2 + (size_t)(t + 1) * BLOCK1 + tid, st_tg + nbuf + tid);
            async_ld_b64(gd2 + (size_t)(t + 1) * BLOCK1 + tid, st_gd + nbuf + tid);
            wait_async<2>();
        } else {
            wait_async<0>();
        }
        float2 v = st_tg[buf + tid];
        int2   g = st_gd[buf + tid];
        atomicMax(&lm[g.x & (G_GROUPS - 1)], __float_as_int(v.x));
        atomicMax(&lm[g.y & (G_GROUPS - 1)], __float_as_int(v.y));
    }
    for (int v = tiles * BLOCK1 + tid; v < nvec2; v += BLOCK1) {   // remainder
        float2 tv = tg2[v]; int2 g = gd2[v];
        atomicMax(&lm[g.x & (G_GROUPS - 1)], __float_as_int(tv.x));
        atomicMax(&lm[g.y & (G_GROUPS - 1)], __float_as_int(tv.y));
    }
#else
    for (int v = tid; v < nvec2; v += BLOCK1) {
        __builtin_prefetch(tg2 + v + BLOCK1, 0, 0);
        __builtin_prefetch(gd2 + v + BLOCK1, 0, 0);
        float2 tv = tg2[v]; int2 g = gd2[v];
        atomicMax(&lm[g.x & (G_GROUPS - 1)], __float_as_int(tv.x));
        atomicMax(&lm[g.y & (G_GROUPS - 1)], __float_as_int(tv.y));
    }
#endif
    __syncthreads();

    float* om = pmax + (size_t)blockIdx.x * G_GROUPS;
#if HAVE_ASYNC_LDS
    float4* om4 = reinterpret_cast<float4*>(om);
    float4* lm4 = reinterpret_cast<float4*>(lm);   // raw bits, read back as float later
    for (int i = tid; i < G_GROUPS / 4; i += BLOCK1) async_st_b128(om4 + i, lm4 + i);
    wait_async<0>();
#else
    for (int i = tid; i < G_GROUPS; i += BLOCK1) om[i] = __int_as_float(lm[i]);
#endif
}

// ---------------------------------------------------------------------------
// Kernel 3: combine partials across chunks, compute per-group loss, reduce.
// ---------------------------------------------------------------------------
__global__ void __launch_bounds__(CBLOCK)
k_combine(const float* __restrict__ psum, const unsigned* __restrict__ pcnt,
          const float* __restrict__ pmax, float* __restrict__ accum, int chunks) {
    const int g = blockIdx.x * CBLOCK + threadIdx.x;   // grid covers exactly G

    float    s = 0.0f;
    unsigned c = 0u;
    float    m = -INFINITY;
    for (int ch = 0; ch < chunks; ++ch) {
        const size_t idx = (size_t)ch * G_GROUPS + g;
        s += psum[idx];
        c += pcnt[idx];
        m  = fmaxf(m, pmax[idx]);
    }

    const float mean = s / (float)c;   // 0/0 -> NaN like the reference for empty groups
    float loss = POS_W * m * log_sigmoid_f(mean)
               + (1.0f - m) * log_sigmoid_f(1.0f - mean);

    for (int off = 16; off > 0; off >>= 1) loss += __shfl_down(loss, off, 32);

    __shared__ float wsum[CBLOCK / 32];
    const int lane = threadIdx.x & 31;
    const int wave = threadIdx.x >> 5;
    if (lane == 0) wsum[wave] = loss;
    __syncthreads();
    if (wave == 0) {
        float v = (lane < (CBLOCK / 32)) ? wsum[lane] : 0.0f;
        for (int off = 16; off > 0; off >>= 1) v += __shfl_down(v, off, 32);
        if (lane == 0) atomicAdd(accum, v);   // global_atomic_add_f32
    }
}

// ---------------------------------------------------------------------------
// Kernel 4: finalize scalar.
// ---------------------------------------------------------------------------
__global__ void k_finalize(const float* __restrict__ accum, float* __restrict__ out) {
    if (threadIdx.x == 0) out[0] = -accum[0] / (float)G_GROUPS;
}

// ---------------------------------------------------------------------------
extern "C" void kernel_launch(void* const* d_in, const int* in_sizes, int n_in,
                              void* d_out, int out_size, void* d_ws, size_t ws_size,
                              hipStream_t stream) {
    const float* inp = (const float*)d_in[0];
    const float* tgt = (const float*)d_in[1];
    const int*   gid = (const int*)d_in[2];
    float*       out = (float*)d_out;
    const int    n   = in_sizes[0];           // 16777216 = 2^24

    // Adaptive chunk count: CHUNKS*G*12 bytes of partials must fit in d_ws.
    size_t per_chunk_bytes = (size_t)G_GROUPS * 12;
    int chunks = 256;
    while (chunks > 1 && ((size_t)chunks * per_chunk_bytes + 256) > ws_size) chunks >>= 1;
    while (chunks > 1 && ((n / 2) % chunks) != 0) chunks >>= 1;  // exact coverage in float2s

    char*     w     = (char*)d_ws;
    float*    psum  = (float*)w;
    unsigned* pcnt  = (unsigned*)(w + (size_t)chunks * G_GROUPS * 4);
    float*    pmax  = (float*)(w + (size_t)chunks * G_GROUPS * 8);
    float*    accum = (float*)(w + (size_t)chunks * G_GROUPS * 12);

    // Allow >64KB dynamic LDS (CDNA5 WGP has 320KB). Not a stream op; capture-safe.
    // NOTE: sizes are macro-fixed (independent of device-only __has_builtin state).
    (void)hipFuncSetAttribute(reinterpret_cast<const void*>(k_sum_count),
                              hipFuncAttributeMaxDynamicSharedMemorySize, SH_SUMCOUNT);
    (void)hipFuncSetAttribute(reinterpret_cast<const void*>(k_max),
                              hipFuncAttributeMaxDynamicSharedMemorySize, SH_MAX);

    const int nvec2 = (n / 2) / chunks;

    hipLaunchKernelGGL(k_init, dim3(1), dim3(32), 0, stream, accum);
    hipLaunchKernelGGL(k_sum_count, dim3(chunks), dim3(BLOCK1), SH_SUMCOUNT, stream,
                       inp, gid, psum, pcnt, nvec2);
    hipLaunchKernelGGL(k_max, dim3(chunks), dim3(BLOCK1), SH_MAX, stream,
                       tgt, gid, pmax, nvec2);
    hipLaunchKernelGGL(k_combine, dim3(G_GROUPS / CBLOCK), dim3(CBLOCK), 0, stream,
                       psum, pcnt, pmax, accum, chunks);
    hipLaunchKernelGGL(k_finalize, dim3(1), dim3(32), 0, stream, accum, out);
}